// PrototypeConsistentLearning_7937099563445
// MI455X (gfx1250) — compile-verified
//
#include <hip/hip_runtime.h>
#include <math.h>

#define B_N 8192
#define D_N 256
#define P_N 4096
#define INV_TEMP 2.0f
#define MOM 0.9f
#define NSPLIT 4
#define NTILES (P_N / 16 / NSPLIT)  // 64 column-tiles of 16 per split

typedef float v2f __attribute__((ext_vector_type(2)));
typedef float v8f __attribute__((ext_vector_type(8)));
typedef unsigned int u32x4 __attribute__((ext_vector_type(4)));
typedef int i32x4 __attribute__((ext_vector_type(4)));
typedef int i32x8 __attribute__((ext_vector_type(8)));

#if __has_builtin(__builtin_amdgcn_tensor_load_to_lds)
#define USE_TDM 1
#else
#define USE_TDM 0
#endif

// ---------------------------------------------------------------------------
// Kernel 1: L2 row normalization. One wave (32 lanes) per row, 8 floats/lane.
// ---------------------------------------------------------------------------
__global__ __launch_bounds__(256) void l2norm_kernel(
    const float* __restrict__ emb, const float* __restrict__ proto,
    float* __restrict__ embN, float* __restrict__ protoN) {
  int gwave = (blockIdx.x * 256 + threadIdx.x) >> 5;
  int lane = threadIdx.x & 31;
  const float* src;
  float* dst;
  if (gwave < B_N) {
    src = emb + (size_t)gwave * D_N;
    dst = embN + (size_t)gwave * D_N;
  } else {
    int r = gwave - B_N;
    if (r >= P_N) return;
    src = proto + (size_t)r * D_N;
    dst = protoN + (size_t)r * D_N;
  }
  float4 a = *(const float4*)(src + lane * 4);
  float4 b = *(const float4*)(src + 128 + lane * 4);
  float ss = a.x * a.x + a.y * a.y + a.z * a.z + a.w * a.w +
             b.x * b.x + b.y * b.y + b.z * b.z + b.w * b.w;
#pragma unroll
  for (int off = 1; off < 32; off <<= 1) ss += __shfl_xor(ss, off, 32);
  float inv = 1.0f / fmaxf(sqrtf(ss), 1e-12f);
  a.x *= inv; a.y *= inv; a.z *= inv; a.w *= inv;
  b.x *= inv; b.y *= inv; b.z *= inv; b.w *= inv;
  *(float4*)(dst + lane * 4) = a;
  *(float4*)(dst + 128 + lane * 4) = b;
}

#if USE_TDM
// Issue a TDM load of a [16 rows x 256 f32] tile of protoN into LDS with a
// 4-DWORD pad after every 256 DWORDs (-> LDS row stride 260 floats).
// D# layout per CDNA5 ISA ch.7/8 (group0 128b, group1 256b; groups 2/3 unused
// for 2D tiles). This toolchain exposes the 6-arg builtin (extra i32x8 group).
__device__ __forceinline__ void tdm_issue_tile(const float* gsrc, void* ldst) {
  unsigned long long ga = (unsigned long long)(uintptr_t)gsrc;
  unsigned int lds_off = (unsigned int)(uintptr_t)ldst;  // addr[31:0] = LDS off
  u32x4 g0;
  g0[0] = 1u;                                   // count=1, user descriptor
  g0[1] = lds_off;                              // lds_addr
  g0[2] = (unsigned int)(ga & 0xffffffffull);   // global_addr[31:0]
  g0[3] = (unsigned int)((ga >> 32) & 0x1ffffffull) | (2u << 30);  // type=2
  i32x8 g1;
  // data_size=4B (2<<16) | pad_enable (1<<20) | pad_interval=256dw (7<<22)
  // | pad_amount=4dw (3<<25)
  g1[0] = (int)((2u << 16) | (1u << 20) | (7u << 22) | (3u << 25));
  g1[1] = (int)(256u << 16);                    // tensor_dim0[15:0] = 256
  g1[2] = (int)(4096u << 16);                   // dim0 hi=0 | tensor_dim1 lo
  g1[3] = (int)(256u << 16);                    // dim1 hi=0 | tile_dim0 = 256
  g1[4] = 16;                                   // tile_dim1 = 16, tile_dim2=0
  g1[5] = 256;                                  // tensor_dim0_stride = 256
  g1[6] = 0;
  g1[7] = 0;
  i32x4 z4 = {0, 0, 0, 0};
  i32x8 z8 = {0, 0, 0, 0, 0, 0, 0, 0};
  __builtin_amdgcn_tensor_load_to_lds(g0, g1, z4, z4, z8, 0);
}
#endif

// ---------------------------------------------------------------------------
// Kernel 2: fused sim GEMM + online masked logsumexp, N-split for occupancy.
// Grid = (B_N/128) * NSPLIT blocks; block = 8 waves, each wave owns 16 rows.
// Each block sweeps P_N/NSPLIT prototype columns in 16-wide tiles staged in
// LDS (stride 260 floats). TDM double-buffers the tiles when available.
// Emits per-(row, split) partial LSE state (max, sum, pos) to workspace.
// ---------------------------------------------------------------------------
__global__ __launch_bounds__(256) void simlse_kernel(
    const float* __restrict__ embN, const float* __restrict__ protoN,
    const int* __restrict__ cid, float* __restrict__ pm,
    float* __restrict__ ps, float* __restrict__ pp) {
  __shared__ float ldsB[2][16 * 260];

  int tid = threadIdx.x;
  int wave = tid >> 5;
  int lane = tid & 31;
  int lane16 = lane & 15;
  int half = lane >> 4;   // 0: C rows r, 1: C rows r+8
  int kb = half * 2;      // K sub-offset of f32 A/B fragments
  int rowblk = blockIdx.x >> 2;          // which 128-row strip
  int nsplit = blockIdx.x & (NSPLIT - 1);
  int tilebase = nsplit * NTILES;        // first 16-col tile of this split
  int rowbase = rowblk * 128 + wave * 16;
  int mrow = rowbase + lane16;

  // Preload this wave's A fragments: A[m][k], k = kc*4 + kb + {0,1}
  v2f aF[64];
#pragma unroll
  for (int kc = 0; kc < 64; ++kc)
    aF[kc] = *(const v2f*)(embN + (size_t)mrow * D_N + kc * 4 + kb);

  int cids[8];
#pragma unroll
  for (int r = 0; r < 8; ++r) cids[r] = cid[rowbase + r + 8 * half];

  float mrun[8], srun[8], pos[8];
#pragma unroll
  for (int r = 0; r < 8; ++r) {
    mrun[r] = -INFINITY; srun[r] = 0.0f; pos[r] = -INFINITY;
  }

#if USE_TDM
  if (wave == 0)  // prologue: stage tile 0 into buffer 0
    tdm_issue_tile(protoN + (size_t)(tilebase * 16) * D_N, &ldsB[0][0]);
#else
  int ldr = tid >> 4;
  int ldc = (tid & 15) * 16;
#endif

  for (int tt = 0; tt < NTILES; ++tt) {
    int p0 = (tilebase + tt) * 16;
#if USE_TDM
    if (tt + 1 < NTILES) {
      if (wave == 0)
        tdm_issue_tile(protoN + (size_t)(p0 + 16) * D_N,
                       &ldsB[(tt + 1) & 1][0]);
      __builtin_amdgcn_s_wait_tensorcnt(1);  // tile tt has landed
    } else {
      __builtin_amdgcn_s_wait_tensorcnt(0);  // last tile has landed
    }
    __syncthreads();
    const float* bBase = &ldsB[tt & 1][lane16 * 260 + kb];
#else
    {
      const float* src = protoN + (size_t)(p0 + ldr) * D_N + ldc;
      float* dst = &ldsB[0][ldr * 260 + ldc];
#pragma unroll
      for (int q = 0; q < 4; ++q)
        *(float4*)(dst + q * 4) = *(const float4*)(src + q * 4);
      if (tt + 1 < NTILES)
        __builtin_prefetch(protoN + (size_t)(p0 + 16 + ldr) * D_N + ldc, 0, 1);
    }
    __syncthreads();
    const float* bBase = &ldsB[0][lane16 * 260 + kb];
#endif

    v8f acc = {};
#pragma unroll
    for (int kc = 0; kc < 64; ++kc) {
      v2f b = *(const v2f*)(bBase + kc * 4);
      acc = __builtin_amdgcn_wmma_f32_16x16x4_f32(
          false, aF[kc], false, b, (short)0, acc, false, false);
    }

    int ncol = p0 + lane16;
#pragma unroll
    for (int r = 0; r < 8; ++r) {
      float v = acc[r] * INV_TEMP;
      if (ncol == cids[r]) {
        pos[r] = v;  // positive column: record, exclude from LSE
      } else {
        float nm = fmaxf(mrun[r], v);
        srun[r] = srun[r] * __expf(mrun[r] - nm) + __expf(v - nm);
        mrun[r] = nm;
      }
    }
    __syncthreads();
  }

  // Combine the 16 lanes of each half-wave (same rows, disjoint columns).
#pragma unroll
  for (int off = 1; off < 16; off <<= 1) {
#pragma unroll
    for (int r = 0; r < 8; ++r) {
      float om = __shfl_xor(mrun[r], off, 32);
      float os = __shfl_xor(srun[r], off, 32);
      float op = __shfl_xor(pos[r], off, 32);
      float nm = fmaxf(mrun[r], om);
      srun[r] = srun[r] * __expf(mrun[r] - nm) + os * __expf(om - nm);
      mrun[r] = nm;
      pos[r] = fmaxf(pos[r], op);
    }
  }
  if (lane16 == 0) {
#pragma unroll
    for (int r = 0; r < 8; ++r) {
      int row = rowbase + r + 8 * half;
      int idx = row * NSPLIT + nsplit;
      pm[idx] = mrun[r];
      ps[idx] = srun[r];
      pp[idx] = pos[r];
    }
  }
}

// ---------------------------------------------------------------------------
// Kernel 3: merge per-split LSE states + deterministic mean (fixed order).
// ---------------------------------------------------------------------------
__global__ __launch_bounds__(256) void loss_reduce_kernel(
    const float* __restrict__ pm, const float* __restrict__ ps,
    const float* __restrict__ pp, float* __restrict__ out) {
  __shared__ float sh[256];
  int t = threadIdx.x;
  float a = 0.0f;
  for (int row = t; row < B_N; row += 256) {
    float m = -INFINITY, pos = -INFINITY;
#pragma unroll
    for (int s = 0; s < NSPLIT; ++s) {
      m = fmaxf(m, pm[row * NSPLIT + s]);
      pos = fmaxf(pos, pp[row * NSPLIT + s]);
    }
    float sum = 0.0f;
#pragma unroll
    for (int s = 0; s < NSPLIT; ++s)
      sum += ps[row * NSPLIT + s] * __expf(pm[row * NSPLIT + s] - m);
    a += (m + __logf(sum)) - pos;
  }
  sh[t] = a;
  __syncthreads();
#pragma unroll
  for (int off = 128; off > 0; off >>= 1) {
    if (t < off) sh[t] += sh[t + off];
    __syncthreads();
  }
  if (t == 0) out[0] = sh[0] / (float)B_N;
}

// ---------------------------------------------------------------------------
// Kernel 4: EMA prototype update. The serial scan is associative per
// prototype: reverse scan gives new_p = 0.9^k * p + sum 0.1*0.9^rank * e.
// One wave per prototype; ids cached in LDS, scanned as int4 chunks.
// ---------------------------------------------------------------------------
__global__ __launch_bounds__(256) void ema_kernel(
    const float* __restrict__ emb, const float* __restrict__ proto,
    const int* __restrict__ cid, float* __restrict__ outP) {
  __shared__ int ids[B_N];
  int tid = threadIdx.x;
  for (int i = tid; i < B_N; i += 256) ids[i] = cid[i];
  __syncthreads();

  int wave = tid >> 5;
  int lane = tid & 31;
  int p = blockIdx.x * 8 + wave;
  int d = lane * 8;

  float acc[8];
#pragma unroll
  for (int j = 0; j < 8; ++j) acc[j] = 0.0f;
  float w = 1.0f;

  for (int i = B_N - 4; i >= 0; i -= 4) {
    int4 q = *(const int4*)&ids[i];
    int qq[4] = {q.w, q.z, q.y, q.x};  // reverse order within chunk
#pragma unroll
    for (int j = 0; j < 4; ++j) {
      if (qq[j] == p) {
        int row = i + 3 - j;
        const float4* e = (const float4*)(emb + (size_t)row * D_N + d);
        float4 e0 = e[0], e1 = e[1];
        float c = (1.0f - MOM) * w;
        acc[0] += c * e0.x; acc[1] += c * e0.y;
        acc[2] += c * e0.z; acc[3] += c * e0.w;
        acc[4] += c * e1.x; acc[5] += c * e1.y;
        acc[6] += c * e1.z; acc[7] += c * e1.w;
        w *= MOM;
      }
    }
  }
  const float* pr = proto + (size_t)p * D_N + d;
  float* po = outP + (size_t)p * D_N + d;  // only 4-byte aligned: scalar stores
#pragma unroll
  for (int j = 0; j < 8; ++j) po[j] = w * pr[j] + acc[j];
}

// ---------------------------------------------------------------------------
extern "C" void kernel_launch(void* const* d_in, const int* in_sizes, int n_in,
                              void* d_out, int out_size, void* d_ws,
                              size_t ws_size, hipStream_t stream) {
  const float* emb = (const float*)d_in[0];    // [8192, 256] f32
  const float* proto = (const float*)d_in[1];  // [4096, 256] f32
  const int* cid = (const int*)d_in[2];        // [8192] int

  float* loss_out = (float*)d_out;             // d_out[0] = loss
  float* proto_out = (float*)d_out + 1;        // d_out[1..] = new prototypes

  // Workspace layout (floats): embN | protoN | pm | ps | pp
  float* embN = (float*)d_ws;
  float* protoN = embN + (size_t)B_N * D_N;
  float* pm = protoN + (size_t)P_N * D_N;
  float* ps = pm + (size_t)B_N * NSPLIT;
  float* pp = ps + (size_t)B_N * NSPLIT;

  l2norm_kernel<<<(B_N + P_N) / 8, 256, 0, stream>>>(emb, proto, embN, protoN);
  simlse_kernel<<<(B_N / 128) * NSPLIT, 256, 0, stream>>>(embN, protoN, cid,
                                                          pm, ps, pp);
  loss_reduce_kernel<<<1, 256, 0, stream>>>(pm, ps, pp, loss_out);
  ema_kernel<<<P_N / 8, 256, 0, stream>>>(emb, proto, cid, proto_out);
}